// AttentionLayer_41403484733850
// MI455X (gfx1250) — compile-verified
//
#include <hip/hip_runtime.h>
#include <math.h>

typedef __attribute__((ext_vector_type(2))) float v2f;
typedef __attribute__((ext_vector_type(8))) float v8f;

#define H 128
#define NH 4
#define HD 32
#define LDSPAD 132   // 128+4: row stride 132 floats -> lane m hits bank 4m (conflict-free)

// ---------------------------------------------------------------------------
// Kernel 1: fused QKV GEMM.  [N,384] = x[N,128] @ [Wq|Wk|Wv] + [bq|bk|bv]
// grid = (ceil(n/64), 6), block = 128 (4 waves).
// Each wave: 64(M) x 16(N) via 4 accumulators -> each B-fragment pair feeds
// 4 WMMAs (4x fewer global loads per WMMA than a 16x16 tile).
// ---------------------------------------------------------------------------
__global__ void qkv_gemm(const float* __restrict__ x,
                         const float* __restrict__ Wq, const float* __restrict__ bq,
                         const float* __restrict__ Wk, const float* __restrict__ bk,
                         const float* __restrict__ Wv, const float* __restrict__ bv,
                         float* __restrict__ Qo, float* __restrict__ Ko,
                         float* __restrict__ Vo, int n) {
    __shared__ float xs[64 * LDSPAD];
    const int t  = threadIdx.x;
    const int m0 = blockIdx.x * 64;

    // stage 64x128 x-tile into LDS (2048 float4 slots, 16 per thread)
    for (int s = t; s < 2048; s += 128) {
        int r = s >> 5, c4 = s & 31;
        int rr = m0 + r; if (rr >= n) rr = n - 1;
        *(float4*)(xs + r * LDSPAD + c4 * 4) =
            *(const float4*)(x + (size_t)rr * H + c4 * 4);
    }
    __syncthreads();

    const int w = t >> 5, lane = t & 31;
    const int c0  = blockIdx.y * 64 + w * 16;     // global column in [0,384)
    const int mat = c0 >> 7;                      // 0=Q 1=K 2=V
    const int col = c0 & 127;
    const float* Wm = (mat == 0) ? Wq : (mat == 1) ? Wk : Wv;
    const float* bm = (mat == 0) ? bq : (mat == 1) ? bk : bv;
    float*       Om = (mat == 0) ? Qo : (mat == 1) ? Ko : Vo;

    const int mloc  = lane & 15;
    const int ncol  = lane & 15;
    const int khalf = (lane >> 4) * 2;

    v8f acc[4] = {};
    #pragma unroll
    for (int kk = 0; kk < 32; ++kk) {
        const int bk = kk * 4 + khalf;
        v2f b = { Wm[(size_t)bk * H + col + ncol],
                  Wm[(size_t)(bk + 1) * H + col + ncol] };
        #pragma unroll
        for (int sub = 0; sub < 4; ++sub) {
            const float* ap = xs + (sub * 16 + mloc) * LDSPAD + bk;
            v2f a = { ap[0], ap[1] };
            acc[sub] = __builtin_amdgcn_wmma_f32_16x16x4_f32(
                false, a, false, b, (short)0, acc[sub], false, false);
        }
    }

    const float bias = bm[col + ncol];
    const int mbase = (lane >> 4) * 8;
    #pragma unroll
    for (int sub = 0; sub < 4; ++sub) {
        #pragma unroll
        for (int i = 0; i < 8; ++i) {
            int m = m0 + sub * 16 + mbase + i;
            if (m < n) Om[(size_t)m * H + col + ncol] = acc[sub][i] + bias;
        }
    }
}

// ---------------------------------------------------------------------------
// Kernel 2: init per-call state: m=-inf, denom=0, agg=0
// ---------------------------------------------------------------------------
__global__ void init_state(float* __restrict__ m, float* __restrict__ denom,
                           float* __restrict__ agg, int n) {
    int i = blockIdx.x * blockDim.x + threadIdx.x;
    size_t tot = (size_t)n * H;
    if ((size_t)i < tot) agg[i] = 0.0f;
    if (i < n * NH) { m[i] = -INFINITY; denom[i] = 0.0f; }
}

// ---------------------------------------------------------------------------
// Kernel 3: per-(edge,head) scores + segment max (native f32 atomic max)
// ---------------------------------------------------------------------------
__global__ void edge_scores(const int* __restrict__ ei,
                            const float* __restrict__ Q, const float* __restrict__ K,
                            float* __restrict__ scores, float* __restrict__ mbuf,
                            int E) {
    int t = blockIdx.x * blockDim.x + threadIdx.x;
    if (t >= E * NH) return;
    int e = t >> 2, h = t & 3;
    int row = ei[e], col = ei[E + e];
    const float4* q = (const float4*)(Q + (size_t)row * H + h * HD);
    const float4* k = (const float4*)(K + (size_t)col * H + h * HD);
    float s = 0.0f;
    #pragma unroll
    for (int i = 0; i < 8; ++i) {
        float4 a = q[i], b = k[i];
        s += a.x * b.x + a.y * b.y + a.z * b.z + a.w * b.w;
    }
    s *= 0.17677669529663687f;  // 1/sqrt(32)
    scores[t] = s;
    atomicMax(mbuf + (size_t)row * NH + h, s);
}

// ---------------------------------------------------------------------------
// Kernel 4: e = exp(s - m[row]); denom += e
// ---------------------------------------------------------------------------
__global__ void edge_exp(const int* __restrict__ ei,
                         float* __restrict__ scores,
                         const float* __restrict__ mbuf,
                         float* __restrict__ denom, int E) {
    int t = blockIdx.x * blockDim.x + threadIdx.x;
    if (t >= E * NH) return;
    int e = t >> 2, h = t & 3;
    int row = ei[e];
    float ew = __expf(scores[t] - mbuf[(size_t)row * NH + h]);
    scores[t] = ew;
    atomicAdd(denom + (size_t)row * NH + h, ew);
}

// ---------------------------------------------------------------------------
// Kernel 5: agg[row] += (e/denom[row]) * V[col]
// ---------------------------------------------------------------------------
__global__ void edge_agg(const int* __restrict__ ei,
                         const float* __restrict__ scores,
                         const float* __restrict__ denom,
                         const float* __restrict__ V,
                         float* __restrict__ agg, int E) {
    int t = blockIdx.x * blockDim.x + threadIdx.x;
    if (t >= E * NH) return;
    int e = t >> 2, h = t & 3;
    int row = ei[e], col = ei[E + e];
    float wgt = scores[t] / denom[(size_t)row * NH + h];
    const float4* v = (const float4*)(V + (size_t)col * H + h * HD);
    float* dst = agg + (size_t)row * H + h * HD;
    #pragma unroll
    for (int i = 0; i < 8; ++i) {
        float4 m4 = v[i];
        atomicAdd(dst + i * 4 + 0, wgt * m4.x);
        atomicAdd(dst + i * 4 + 1, wgt * m4.y);
        atomicAdd(dst + i * 4 + 2, wgt * m4.z);
        atomicAdd(dst + i * 4 + 3, wgt * m4.w);
    }
}

// ---------------------------------------------------------------------------
// Kernel 6: hn = LayerNorm(ReLU([x|agg] @ W1 + b1)) * g + b
// grid = ceil(n/64), block = 256 (8 waves x 16 cols = 128 cols), K=256 in
// 2 LDS phases; each wave: 64x16 via 4 accumulators. The x-tile LDS is
// reused (barrier-protected) for the ReLU tile before LayerNorm.
// ---------------------------------------------------------------------------
__global__ void mlp1_ln(const float* __restrict__ x, const float* __restrict__ agg,
                        const float* __restrict__ W1, const float* __restrict__ b1,
                        const float* __restrict__ g, const float* __restrict__ be,
                        float* __restrict__ hn, int n) {
    __shared__ float xs[64 * LDSPAD];
    const int t  = threadIdx.x;
    const int m0 = blockIdx.x * 64;
    const int w = t >> 5, lane = t & 31;
    const int c0 = w * 16;
    const int mloc = lane & 15, ncol = lane & 15, khalf = (lane >> 4) * 2;

    v8f acc[4] = {};
    for (int p = 0; p < 2; ++p) {
        const float* src = p ? agg : x;
        __syncthreads();
        for (int s = t; s < 2048; s += 256) {
            int r = s >> 5, c4 = s & 31;
            int rr = m0 + r; if (rr >= n) rr = n - 1;
            *(float4*)(xs + r * LDSPAD + c4 * 4) =
                *(const float4*)(src + (size_t)rr * H + c4 * 4);
        }
        __syncthreads();
        #pragma unroll
        for (int kk = 0; kk < 32; ++kk) {
            const int bk = kk * 4 + khalf;
            const int gk = p * 128 + bk;
            v2f b = { W1[(size_t)gk * H + c0 + ncol],
                      W1[(size_t)(gk + 1) * H + c0 + ncol] };
            #pragma unroll
            for (int sub = 0; sub < 4; ++sub) {
                const float* ap = xs + (sub * 16 + mloc) * LDSPAD + bk;
                v2f a = { ap[0], ap[1] };
                acc[sub] = __builtin_amdgcn_wmma_f32_16x16x4_f32(
                    false, a, false, b, (short)0, acc[sub], false, false);
            }
        }
    }

    // ReLU -> reuse xs as the 64x128 hidden tile
    __syncthreads();
    const int mbase = (lane >> 4) * 8;
    const float bias = b1[c0 + ncol];
    #pragma unroll
    for (int sub = 0; sub < 4; ++sub) {
        #pragma unroll
        for (int i = 0; i < 8; ++i) {
            float v = acc[sub][i] + bias;
            xs[(sub * 16 + mbase + i) * LDSPAD + c0 + ncol] = fmaxf(v, 0.0f);
        }
    }
    __syncthreads();

    // LayerNorm: 4 threads per row (width-4 shfl reduction), 32 cols each
    const int r = t >> 2, j = t & 3;
    float s = 0.0f, sq = 0.0f;
    #pragma unroll
    for (int i = 0; i < 32; ++i) {
        float v = xs[r * LDSPAD + j * 32 + i];
        s += v; sq += v * v;
    }
    s  += __shfl_xor(s,  1, 4);  s  += __shfl_xor(s,  2, 4);
    sq += __shfl_xor(sq, 1, 4);  sq += __shfl_xor(sq, 2, 4);
    const float mu  = s * (1.0f / 128.0f);
    const float var = sq * (1.0f / 128.0f) - mu * mu;
    const float rs  = rsqrtf(var + 1e-5f);
    if (m0 + r < n) {
        #pragma unroll
        for (int i = 0; i < 32; ++i) {
            int c = j * 32 + i;
            float v = xs[r * LDSPAD + c];
            hn[(size_t)(m0 + r) * H + c] = (v - mu) * rs * g[c] + be[c];
        }
    }
}

// ---------------------------------------------------------------------------
// Kernel 7: out = hn @ W2 + b2 + x    grid = (ceil(n/64), 2), block = 128
// ---------------------------------------------------------------------------
__global__ void mlp2(const float* __restrict__ hn, const float* __restrict__ W2,
                     const float* __restrict__ b2, const float* __restrict__ x,
                     float* __restrict__ out, int n) {
    __shared__ float xs[64 * LDSPAD];
    const int t  = threadIdx.x;
    const int m0 = blockIdx.x * 64;
    for (int s = t; s < 2048; s += 128) {
        int r = s >> 5, c4 = s & 31;
        int rr = m0 + r; if (rr >= n) rr = n - 1;
        *(float4*)(xs + r * LDSPAD + c4 * 4) =
            *(const float4*)(hn + (size_t)rr * H + c4 * 4);
    }
    __syncthreads();

    const int w = t >> 5, lane = t & 31;
    const int c0 = blockIdx.y * 64 + w * 16;
    const int mloc = lane & 15, ncol = lane & 15, khalf = (lane >> 4) * 2;

    v8f acc[4] = {};
    #pragma unroll
    for (int kk = 0; kk < 32; ++kk) {
        const int bk = kk * 4 + khalf;
        v2f b = { W2[(size_t)bk * H + c0 + ncol],
                  W2[(size_t)(bk + 1) * H + c0 + ncol] };
        #pragma unroll
        for (int sub = 0; sub < 4; ++sub) {
            const float* ap = xs + (sub * 16 + mloc) * LDSPAD + bk;
            v2f a = { ap[0], ap[1] };
            acc[sub] = __builtin_amdgcn_wmma_f32_16x16x4_f32(
                false, a, false, b, (short)0, acc[sub], false, false);
        }
    }
    const float bias = b2[c0 + ncol];
    const int mbase = (lane >> 4) * 8;
    #pragma unroll
    for (int sub = 0; sub < 4; ++sub) {
        #pragma unroll
        for (int i = 0; i < 8; ++i) {
            int m = m0 + sub * 16 + mbase + i;
            if (m < n)
                out[(size_t)m * H + c0 + ncol] =
                    acc[sub][i] + bias + x[(size_t)m * H + c0 + ncol];
        }
    }
}

// ---------------------------------------------------------------------------
extern "C" void kernel_launch(void* const* d_in, const int* in_sizes, int n_in,
                              void* d_out, int out_size, void* d_ws, size_t ws_size,
                              hipStream_t stream) {
    const float* x    = (const float*)d_in[0];
    const int*   ei   = (const int*)  d_in[1];   // [2,E] (JAX default: int32)
    const float* Wq   = (const float*)d_in[2];
    const float* bq   = (const float*)d_in[3];
    const float* Wk   = (const float*)d_in[4];
    const float* bk   = (const float*)d_in[5];
    const float* Wv   = (const float*)d_in[6];
    const float* bv   = (const float*)d_in[7];
    const float* W1   = (const float*)d_in[8];
    const float* b1   = (const float*)d_in[9];
    const float* ln_g = (const float*)d_in[10];
    const float* ln_b = (const float*)d_in[11];
    const float* W2   = (const float*)d_in[12];
    const float* b2   = (const float*)d_in[13];
    float* out = (float*)d_out;

    const int n = in_sizes[0] / H;       // 50000
    const int E = in_sizes[1] / 2;       // 800000

    // workspace layout (floats)
    float* ws = (float*)d_ws;
    const size_t NHF = (size_t)n * H;
    float* Qb     = ws;
    float* Kb     = Qb + NHF;
    float* Vb     = Kb + NHF;
    float* agg    = Vb + NHF;
    float* hn     = agg + NHF;
    float* scores = hn + NHF;                 // E*NH
    float* mbuf   = scores + (size_t)E * NH;  // n*NH
    float* denom  = mbuf + (size_t)n * NH;    // n*NH

    const int mtiles64 = (n + 63) / 64;

    qkv_gemm<<<dim3(mtiles64, 6), 128, 0, stream>>>(x, Wq, bq, Wk, bk, Wv, bv,
                                                    Qb, Kb, Vb, n);

    const int initThreads = 256;
    const int initBlocks  = (int)(((size_t)n * H + initThreads - 1) / initThreads);
    init_state<<<initBlocks, initThreads, 0, stream>>>(mbuf, denom, agg, n);

    const int eThreads = 256;
    const int eBlocks  = (E * NH + eThreads - 1) / eThreads;
    edge_scores<<<eBlocks, eThreads, 0, stream>>>(ei, Qb, Kb, scores, mbuf, E);
    edge_exp   <<<eBlocks, eThreads, 0, stream>>>(ei, scores, mbuf, denom, E);
    edge_agg   <<<eBlocks, eThreads, 0, stream>>>(ei, scores, denom, Vb, agg, E);

    mlp1_ln<<<mtiles64, 256, 0, stream>>>(x, agg, W1, b1, ln_g, ln_b, hn, n);
    mlp2   <<<dim3(mtiles64, 2), 128, 0, stream>>>(hn, W2, b2, x, out, n);

    (void)n_in; (void)out_size; (void)ws_size;
}